// AnalyiticLinearLocallyConnected_1889785610311
// MI455X (gfx1250) — compile-verified
//
#include <hip/hip_runtime.h>

typedef float v2f __attribute__((ext_vector_type(2)));
typedef float v8f __attribute__((ext_vector_type(8)));

#define E_ 8
#define N_ 8192
#define D_ 128
#define BETA 0.01f

// -----------------------------------------------------------------------------
// Kernel 1: per expert, S = xs^T xs and C = xs^T dxs  (both D x D, K = N)
// 1024-thread blocks (32 waves = WGP max). Wave (kc,jt): kc in 0..3 splits the
// K(=N) range, jt in 0..7 picks the j-tile; all waves share the block's i-tile.
// Each wave accumulates a 16x16 tile of S and C with V_WMMA_F32_16X16X4_F32
// over N/4 samples; partials are combined with a deterministic LDS reduction.
// -----------------------------------------------------------------------------
__global__ __launch_bounds__(1024) void gram_kernel(const float* __restrict__ xs,
                                                    const float* __restrict__ dxs,
                                                    float* __restrict__ S,
                                                    float* __restrict__ C) {
  const int lane = threadIdx.x & 31;
  const int wave = threadIdx.x >> 5;     // 0..31
  const int jt   = wave & 7;             // j-tile
  const int kc   = wave >> 3;            // K-chunk 0..3
  const int e    = blockIdx.x >> 3;      // 8 blocks per expert
  const int ti   = blockIdx.x & 7;       // i-tile shared by whole block
  const int m    = lane & 15;
  const int half = lane >> 4;            // 0: holds K=0,1  1: K=2,3

  const float* xe  = xs  + (size_t)e * N_ * D_;
  const float* dxe = dxs + (size_t)e * N_ * D_;
  const int ia = ti * 16 + m;            // A column (row of S)
  const int ja = jt * 16 + m;            // B column (col of S)

  v8f accS = {};
  v8f accC = {};
  const int nBeg = kc * (N_ / 4);
  const int nEnd = nBeg + (N_ / 4);
  for (int n0 = nBeg; n0 < nEnd; n0 += 4) {
    const int k0 = n0 + 2 * half;
    v2f a, b, b2;
    // A-matrix 16x4 f32 layout: v0 = K(2*half), v1 = K(2*half+1), M = lane%16
    a.x  = xe [(size_t)(k0    ) * D_ + ia];
    a.y  = xe [(size_t)(k0 + 1) * D_ + ia];
    // B-matrix 4x16 f32 layout mirrors A: rows K, cols N = lane%16
    b.x  = xe [(size_t)(k0    ) * D_ + ja];
    b.y  = xe [(size_t)(k0 + 1) * D_ + ja];
    b2.x = dxe[(size_t)(k0    ) * D_ + ja];
    b2.y = dxe[(size_t)(k0 + 1) * D_ + ja];
    accS = __builtin_amdgcn_wmma_f32_16x16x4_f32(false, a, false, b,
                                                 (short)0, accS, false, false);
    accC = __builtin_amdgcn_wmma_f32_16x16x4_f32(false, a, false, b2,
                                                 (short)0, accC, false, false);
  }

  // Deterministic K-chunk reduction in LDS: waves kc=1..3 park their partials,
  // waves kc=0 accumulate them in a fixed order.
  __shared__ float red[24 * 512];        // 48 KB: 24 waves x (256 S + 256 C)
  if (kc != 0) {
    float* myS = &red[(wave - 8) * 512 + lane * 8];
#pragma unroll
    for (int v = 0; v < 8; ++v) {
      myS[v]       = accS[v];
      myS[256 + v] = accC[v];
    }
  }
  __syncthreads();
  if (kc == 0) {
#pragma unroll
    for (int c = 1; c < 4; ++c) {
      const float* oS = &red[(((c << 3) | jt) - 8) * 512 + lane * 8];
#pragma unroll
      for (int v = 0; v < 8; ++v) {
        accS[v] += oS[v];
        accC[v] += oS[256 + v];
      }
    }
    float* Se = S + (size_t)e * D_ * D_;
    float* Ce = C + (size_t)e * D_ * D_;
#pragma unroll
    for (int v = 0; v < 8; ++v) {
      const int row = ti * 16 + v + 8 * half;  // C/D layout: M = v + 8*(lane>=16)
      const int col = jt * 16 + m;
      Se[row * D_ + col] = accS[v];
      Ce[row * D_ + col] = accC[v];
    }
  }
}

// -----------------------------------------------------------------------------
// Kernel 2: for each (e,p) solve (g_p g_p^T o S + beta*I) w = g_p o C[:,p].
// A is SPD (Schur product of PSD matrices + beta*I, diag >= beta): Gauss-Jordan
// without pivoting is stable. One block per system; 256 threads = 2 per row
// with stride-2 column interleave. Column k is dead after step k, so it is
// never written -> no intra-row race on the multiplier read A[r][k].
// A lives in dynamic LDS (128x129 f32 padded) -> ~66 KB of the 320 KB WGP LDS.
// -----------------------------------------------------------------------------
__global__ __launch_bounds__(256) void solve_kernel(const float* __restrict__ S,
                                                    const float* __restrict__ C,
                                                    const int* __restrict__ G,
                                                    float* __restrict__ W) {
  extern __shared__ float sm[];
  float* A   = sm;                 // 128 x 129 (padded to dodge bank conflicts)
  float* gsh = sm + 128 * 129;     // 128: mask column g_p
  float* bv  = gsh + 128;          // 128: rhs
  float* aux = bv + 128;           // [0] = 1/pivot

  const int tid = threadIdx.x;
  const int r   = tid >> 1;        // row index
  const int h   = tid & 1;         // column interleave phase
  const int e   = blockIdx.x >> 7;
  const int p   = blockIdx.x & 127;

  const float* Se = S + (size_t)e * D_ * D_;
  const float* Ce = C + (size_t)e * D_ * D_;
  const int*   Ge = G + (size_t)e * D_ * D_;

  if (h == 0) gsh[r] = (float)Ge[r * D_ + p];   // gp[e,p,r] = G[e,r,p]
  __syncthreads();

  const float gr = gsh[r];
  for (int j = h; j < D_; j += 2) {
    float v = gr * gsh[j] * Se[r * D_ + j];
    if (j == r) v += BETA;
    A[r * 129 + j] = v;
  }
  if (h == 0) bv[r] = gr * Ce[r * D_ + p];
  __syncthreads();

  for (int k = 0; k < D_; ++k) {
    if (r == k && h == 0) aux[0] = 1.0f / A[k * 129 + k];
    __syncthreads();
    if (r != k) {
      const float f = A[r * 129 + k] * aux[0];
      for (int j = k + 1 + h; j < D_; j += 2)   // column k never rewritten
        A[r * 129 + j] -= f * A[k * 129 + j];
      if (h == 0) bv[r] -= f * bv[k];
    }
    __syncthreads();
  }

  // Live part of A is now diagonal.
  if (h == 0) W[(size_t)e * D_ * D_ + p * D_ + r] = bv[r] / A[r * 129 + r];
}

// -----------------------------------------------------------------------------
// Kernel 3: y[e,s,p] = sum_i W[e,p,i] * xs[e,s,i]   (N x D GEMM with K = D)
// One wave = one 16x16 tile of y, K-loop of 32 WMMA f32 16x16x4 steps.
// A/B lane pairs are contiguous -> compiler merges them into global_load_b64.
// -----------------------------------------------------------------------------
__global__ __launch_bounds__(256) void apply_kernel(const float* __restrict__ xs,
                                                    const float* __restrict__ W,
                                                    float* __restrict__ y) {
  const int lane = threadIdx.x & 31;
  const int wave = threadIdx.x >> 5;
  const int e    = blockIdx.x >> 9;                    // 512 blocks per expert
  const int tile = ((blockIdx.x & 511) << 3) | wave;   // 0..4095
  const int st   = tile >> 3;                          // s-tile 0..511
  const int pt   = tile & 7;                           // p-tile 0..7
  const int m    = lane & 15;
  const int half = lane >> 4;

  const float* xe = xs + (size_t)e * N_ * D_;
  const float* We = W  + (size_t)e * D_ * D_;
  const int srow = st * 16 + m;    // A: row s
  const int prow = pt * 16 + m;    // B column p -> row p of W

  v8f acc = {};
#pragma unroll
  for (int i0 = 0; i0 < D_; i0 += 4) {
    const int k0 = i0 + 2 * half;
    v2f a, b;
    a.x = xe[(size_t)srow * D_ + k0];
    a.y = xe[(size_t)srow * D_ + k0 + 1];
    b.x = We[(size_t)prow * D_ + k0];       // B[k,p] = W[p, i0+k]
    b.y = We[(size_t)prow * D_ + k0 + 1];
    acc = __builtin_amdgcn_wmma_f32_16x16x4_f32(false, a, false, b,
                                                (short)0, acc, false, false);
  }

  float* ye = y + (size_t)e * N_ * D_;
#pragma unroll
  for (int v = 0; v < 8; ++v) {
    const int row = st * 16 + v + 8 * half;
    const int col = pt * 16 + m;
    ye[(size_t)row * D_ + col] = acc[v];
  }
}

// -----------------------------------------------------------------------------
extern "C" void kernel_launch(void* const* d_in, const int* in_sizes, int n_in,
                              void* d_out, int out_size, void* d_ws, size_t ws_size,
                              hipStream_t stream) {
  (void)in_sizes; (void)n_in; (void)out_size; (void)ws_size;

  const float* x  = (const float*)d_in[0];   // (E,N,1,D) contiguous == (E,N,D)
  const float* dx = (const float*)d_in[1];   // (E,N,D,1) contiguous == (E,N,D)
  const int*   G  = (const int*)d_in[2];     // (E,D,D)
  float* y = (float*)d_out;                  // (E,N,D) fp32

  const size_t EDD = (size_t)E_ * D_ * D_;
  float* S = (float*)d_ws;        // E*D*D
  float* C = S + EDD;             // E*D*D
  float* W = C + EDD;             // E*D*D   (w[e,p,i])

  gram_kernel<<<E_ * 8, 1024, 0, stream>>>(x, dx, S, C);

  const size_t shmem = (size_t)(128 * 129 + 128 + 128 + 1) * sizeof(float);
  solve_kernel<<<E_ * D_, 256, shmem, stream>>>(S, C, G, W);

  apply_kernel<<<E_ * 512, 256, 0, stream>>>(x, W, y);
}